// CenterLoss_24575802868410
// MI455X (gfx1250) — compile-verified
//
#include <hip/hip_runtime.h>

// CenterLoss on gfx1250 (MI455X).
//
// Key algebraic optimization: distmat*onehot + clip means we only ever need
// distmat[i, labels[i]]; everything else contributes exactly 1e-12.
//   loss = (1/B) * sum_i clip(||x_i||^2 + ||c_li||^2 - 2<x_i,c_li>) + (C-1)*1e-12
// => O(B*D) gather-reduction (~16 MB traffic, ~0.7us @ 23.3 TB/s) instead of a
//    B x C x D GEMM.
//
// The per-sample dot products are computed on the matrix pipe: each wave owns
// a 16-sample tile and chains V_WMMA_F32_16X16X4_F32 over K=512 (128 steps);
// the 16x16 f32 accumulator's diagonal is the 16 needed dot products.
// A-fragment lane layout (ISA 7.12.2, 32-bit A 16x4): lane L -> row M=L%16,
// K = 2*(L>>4) + {0,1} => one aligned float2 per lane per K-step, and the
// mirrored B layout makes the center-row load the identical pattern.

typedef __attribute__((ext_vector_type(2))) float v2f;
typedef __attribute__((ext_vector_type(8))) float v8f;

#define TILE_M          16
#define WAVES_PER_BLOCK 8
#define BLOCK_THREADS   (WAVES_PER_BLOCK * 32)

__global__ __launch_bounds__(BLOCK_THREADS)
void center_loss_tile_kernel(const float* __restrict__ x,
                             const float* __restrict__ centers,
                             const int*   __restrict__ labels,
                             float*       __restrict__ tile_sums,
                             int feat_dim)
{
    __shared__ float s_xp[WAVES_PER_BLOCK][32];   // per-lane ||x||^2 partials
    __shared__ float s_cp[WAVES_PER_BLOCK][32];   // per-lane ||c||^2 partials
    __shared__ float s_dg[WAVES_PER_BLOCK][16];   // WMMA diagonal = dot(x_i, c_i)

    const int lane = threadIdx.x & 31;
    const int wv   = threadIdx.x >> 5;
    const int tile = blockIdx.x * WAVES_PER_BLOCK + wv;  // 16 samples per wave

    const int m      = lane & 15;      // matrix row (sample within tile)
    const int half   = lane >> 4;      // K sub-slot: K = 2*half + {0,1} within each 4-wide slice
    const int sample = tile * TILE_M + m;

    const float* row_x = x + (size_t)sample * feat_dim + 2 * half;
    const int    lbl   = labels[sample];                 // JAX default: int32
    const float* row_c = centers + (size_t)lbl * feat_dim + 2 * half;
    __builtin_prefetch(row_c, 0, 0);                     // global_prefetch_b8 on the gathered row

    v8f   acc = {};                                      // 16x16 f32 accumulator
    float xp = 0.f, cp = 0.f;

    // 128 chained f32 WMMAs cover K = 512 (4 K-values per instruction).
    #pragma unroll 4
    for (int kb = 0; kb < feat_dim; kb += 4) {
        v2f a = *(const v2f*)(row_x + kb);               // A[M][kb+2h .. kb+2h+1]
        v2f b = *(const v2f*)(row_c + kb);               // B (= gathered C^T) same pattern
        acc = __builtin_amdgcn_wmma_f32_16x16x4_f32(
            /*neg_a=*/false, a, /*neg_b=*/false, b,
            /*c_mod=*/(short)0, acc, /*reuse_a=*/false, /*reuse_b=*/false);
        xp = fmaf(a.x, a.x, fmaf(a.y, a.y, xp));         // ||x_m||^2 partial (this lane's K subset)
        cp = fmaf(b.x, b.x, fmaf(b.y, b.y, cp));         // ||c_m||^2 partial
    }

    s_xp[wv][lane] = xp;
    s_cp[wv][lane] = cp;

    // Extract accumulator diagonal (C/D layout, ISA 7.12.2):
    //   diag i in [0,7]  -> lane i,      VGPR i
    //   diag i in [8,15] -> lane i+16,   VGPR i-8
    {
        const int di = (lane < 8) ? lane : (lane - 24);  // meaningful for lane<8 or lane>=24
        float dv = acc[0];
        #pragma unroll
        for (int j = 1; j < 8; ++j) dv = (di == j) ? acc[j] : dv;
        if (lane < 8)        s_dg[wv][lane]      = dv;
        else if (lane >= 24) s_dg[wv][lane - 16] = dv;
    }
    __syncthreads();

    float d = 0.f;
    if (lane < 16) {
        const float xsq = s_xp[wv][lane] + s_xp[wv][lane + 16];
        const float csq = s_cp[wv][lane] + s_cp[wv][lane + 16];
        const float dot = s_dg[wv][lane];
        d = xsq + csq - 2.0f * dot;                      // same formula as reference
        d = fminf(fmaxf(d, 1e-12f), 1e12f);              // clip
    }
    // wave32 tree reduction (lanes >=16 contribute 0)
    #pragma unroll
    for (int off = 16; off >= 1; off >>= 1) d += __shfl_xor(d, off, 32);
    if (lane == 0) tile_sums[tile] = d;
}

// Deterministic final reduction (no float atomics -> bitwise-stable replays).
__global__ __launch_bounds__(BLOCK_THREADS)
void center_loss_finalize(const float* __restrict__ tile_sums, int n_tiles,
                          float* __restrict__ out, float inv_batch, float base_term)
{
    __shared__ float red[WAVES_PER_BLOCK];
    float s = 0.f;
    for (int i = threadIdx.x; i < n_tiles; i += BLOCK_THREADS) s += tile_sums[i];
    #pragma unroll
    for (int off = 16; off >= 1; off >>= 1) s += __shfl_xor(s, off, 32);
    const int lane = threadIdx.x & 31, wv = threadIdx.x >> 5;
    if (lane == 0) red[wv] = s;
    __syncthreads();
    if (threadIdx.x == 0) {
        float t = 0.f;
        #pragma unroll
        for (int w = 0; w < WAVES_PER_BLOCK; ++w) t += red[w];
        out[0] = fmaf(t, inv_batch, base_term);
    }
}

extern "C" void kernel_launch(void* const* d_in, const int* in_sizes, int n_in,
                              void* d_out, int out_size, void* d_ws, size_t ws_size,
                              hipStream_t stream)
{
    const float* x       = (const float*)d_in[0];   // [B, D] f32
    const float* centers = (const float*)d_in[1];   // [C, D] f32
    const int*   labels  = (const int*)  d_in[2];   // [B] int32 (JAX x64 disabled)
    float*       out     = (float*)d_out;           // scalar loss

    const int B    = in_sizes[2];                   // 4096
    const int feat = in_sizes[0] / B;               // 512
    const int C    = in_sizes[1] / feat;            // 10000

    const int n_tiles = B / TILE_M;                 // 256 (one wave each)
    const int blocks  = n_tiles / WAVES_PER_BLOCK;  // 32

    float* tile_sums = (float*)d_ws;                // n_tiles floats of scratch

    center_loss_tile_kernel<<<blocks, BLOCK_THREADS, 0, stream>>>(
        x, centers, labels, tile_sums, feat);

    // Every non-label entry of the clipped distmat contributes exactly 1e-12:
    // (B*(C-1)*1e-12)/B = (C-1)*1e-12.
    const float base = (float)((double)(C - 1) * 1e-12);
    center_loss_finalize<<<1, BLOCK_THREADS, 0, stream>>>(
        tile_sums, n_tiles, out, 1.0f / (float)B, base);
}